// RasterizationModelRGBManual_notile_46462956208510
// MI455X (gfx1250) — compile-verified
//
#include <hip/hip_runtime.h>
#include <hip/hip_bf16.h>

typedef float v2f __attribute__((ext_vector_type(2)));
typedef float v8f __attribute__((ext_vector_type(8)));

#define NG     768          // gaussians
#define WPX    224
#define HPX    126
#define NPIX   (WPX*HPX)    // 28224
#define TILES_M (NPIX/16)   // 1764
#define TILES_N (NG/16)     // 48
#define TOTAL_WAVES (TILES_M*TILES_N)  // 84672
#define WAVES_PER_BLOCK 8
#define MAIN_BLOCKS (TOTAL_WAVES/WAVES_PER_BLOCK) // 10584

// ---------------------------------------------------------------------------
// Kernel 1: per-gaussian preprocessing -> v-table [8][NG] (rows 6,7 zero).
// power(p,n) = x^2*v0 + y^2*v1 + xy*v2 + x*v3 + y*v4 + 1*v5   (ln(opac) folded)
// ---------------------------------------------------------------------------
__global__ __launch_bounds__(256) void splat_pre(
    const float* __restrict__ pose,   // [4][4]
    const float* __restrict__ mh,     // [4][NG]  (means_hom transposed)
    const float* __restrict__ cov,    // [NG][3][3]
    const float* __restrict__ opac,   // [NG]
    const float* __restrict__ Kmat,   // [3][3]
    const float* __restrict__ limx,   // [1]
    const float* __restrict__ limy,   // [1]
    const float* __restrict__ fxp,    // [1]
    const float* __restrict__ fyp,    // [1]
    float* __restrict__ vtab)         // [8][NG]
{
    int n = blockIdx.x * blockDim.x + threadIdx.x;
    if (n >= NG) return;

    float hx = mh[0*NG+n], hy = mh[1*NG+n], hz = mh[2*NG+n], hw = mh[3*NG+n];

    // means_cam_hom = pose @ [hx,hy,hz,hw]
    float cam[4];
#pragma unroll
    for (int i = 0; i < 4; ++i)
        cam[i] = pose[i*4+0]*hx + pose[i*4+1]*hy + pose[i*4+2]*hz + pose[i*4+3]*hw;
    float invw = 1.0f / cam[3];
    float xc = cam[0]*invw, yc = cam[1]*invw, z = cam[2]*invw;

    // cov_cam = R * C * R^T, R = pose[:3,:3]
    float R[3][3], C[3][3], RC[3][3], cc[3][3];
#pragma unroll
    for (int i = 0; i < 3; ++i)
#pragma unroll
        for (int j = 0; j < 3; ++j) {
            R[i][j] = pose[i*4+j];
            C[i][j] = cov[n*9 + i*3 + j];
        }
#pragma unroll
    for (int i = 0; i < 3; ++i)
#pragma unroll
        for (int k = 0; k < 3; ++k)
            RC[i][k] = R[i][0]*C[0][k] + R[i][1]*C[1][k] + R[i][2]*C[2][k];
#pragma unroll
    for (int i = 0; i < 3; ++i)
#pragma unroll
        for (int l = 0; l < 3; ++l)
            cc[i][l] = RC[i][0]*R[l][0] + RC[i][1]*R[l][1] + RC[i][2]*R[l][2];

    // projection
    float p0 = Kmat[0]*xc + Kmat[1]*yc + Kmat[2]*z;
    float p1 = Kmat[3]*xc + Kmat[4]*yc + Kmat[5]*z;
    float p2 = Kmat[6]*xc + Kmat[7]*yc + Kmat[8]*z;
    float invp2 = 1.0f / p2;
    float m2x = p0*invp2, m2y = p1*invp2;

    float lx = limx[0], ly = limy[0], fx = fxp[0], fy = fyp[0];
    float invz = 1.0f / z;
    float tx = z * fminf(lx, fmaxf(-lx, xc*invz));
    float ty = z * fminf(ly, fmaxf(-ly, yc*invz));
    float invz2 = invz*invz;
    float J00 = fx*invz, J02 = -fx*tx*invz2;
    float J11 = fy*invz, J12 = -fy*ty*invz2;

    float c00 = J00*J00*cc[0][0] + J00*J02*cc[0][2] + J02*J00*cc[2][0] + J02*J02*cc[2][2] + 0.3f;
    float c11 = J11*J11*cc[1][1] + J11*J12*cc[1][2] + J12*J11*cc[2][1] + J12*J12*cc[2][2] + 0.3f;
    float c01 = J00*J11*cc[0][1] + J00*J12*cc[0][2] + J02*J11*cc[2][1] + J02*J12*cc[2][2];
    float c10 = J11*J00*cc[1][0] + J11*J02*cc[1][2] + J12*J00*cc[2][0] + J12*J02*cc[2][2];

    float inv_det = 1.0f / (c00*c11 - c01*c10);
    float conic00 =  inv_det * c11;
    float conic11 =  inv_det * c00;
    float conic01 = -inv_det * c01;
    float conic10 = -inv_det * c10;

    float A  = -0.5f * conic00;
    float B  = -0.5f * conic11;
    float Cq = -0.5f * (conic01 + conic10);

    vtab[0*NG+n] = A;
    vtab[1*NG+n] = B;
    vtab[2*NG+n] = Cq;
    vtab[3*NG+n] = -2.0f*A*m2x - Cq*m2y;
    vtab[4*NG+n] = -2.0f*B*m2y - Cq*m2x;
    vtab[5*NG+n] = A*m2x*m2x + B*m2y*m2y + Cq*m2x*m2y + __logf(opac[n]);
    vtab[6*NG+n] = 0.0f;   // K padding rows
    vtab[7*NG+n] = 0.0f;
}

// ---------------------------------------------------------------------------
// Kernel 2: [28224 x 8] x [8 x 768] GEMM via V_WMMA_F32_16X16X4_F32 (x2, K=8)
// then alpha_clip = min(exp(power), 0.99). One 16x16 tile per wave.
// ---------------------------------------------------------------------------
__global__ __launch_bounds__(256) void splat_main(
    const float* __restrict__ vtab,   // [8][NG]
    float* __restrict__ out)          // [NPIX][NG]
{
    const int lane = threadIdx.x & 31;
    const int wave = threadIdx.x >> 5;
    const int gw   = blockIdx.x * WAVES_PER_BLOCK + wave;
    const int tileN = gw % TILES_N;
    const int tileM = gw / TILES_N;
    const int half  = lane >> 4;     // 0: lanes 0-15, 1: lanes 16-31
    const int m     = lane & 15;

    // ---- B operands (per-gaussian v-vectors, K-major table) ----
    const int n = tileN * 16 + m;
    v2f b0, b1;
    b0.x = vtab[(half*2 + 0)*NG + n];   // K = 0 / 2
    b0.y = vtab[(half*2 + 1)*NG + n];   // K = 1 / 3
    b1.x = vtab[(half*2 + 4)*NG + n];   // K = 4 / 6(zero)
    b1.y = vtab[(half*2 + 5)*NG + n];   // K = 5 / 7(zero)

    // ---- A operands (pixel features), branch-free for uniform EXEC ----
    const int p  = tileM * 16 + m;      // pixel index for M = m
    const float px = (float)(p % WPX);
    const float py = (float)(p / WPX);
    v2f a0, a1;
    a0.x = (half == 0) ? px*px : px*py;  // K0 = x^2 | K2 = xy
    a0.y = (half == 0) ? py*py : px;     // K1 = y^2 | K3 = x
    a1.x = (half == 0) ? py    : 0.0f;   // K4 = y   | K6 = 0
    a1.y = (half == 0) ? 1.0f  : 0.0f;   // K5 = 1   | K7 = 0

    v8f c = {0.f,0.f,0.f,0.f,0.f,0.f,0.f,0.f};
    c = __builtin_amdgcn_wmma_f32_16x16x4_f32(false, a0, false, b0, (short)0, c, false, false);
    c = __builtin_amdgcn_wmma_f32_16x16x4_f32(false, a1, false, b1, (short)0, c, false, false);

    // ---- epilogue: alpha_clip = min(exp(power), 0.99) ----
    // C layout: VGPR r -> M = r (lanes 0-15) or M = 8+r (lanes 16-31), N = m
    const long rowBase = (long)(tileM * 16 + half * 8) * NG;
#pragma unroll
    for (int r = 0; r < 8; ++r) {
        float val = __expf(c[r]);
        val = fminf(val, 0.99f);
        out[rowBase + (long)r * NG + n] = val;
    }
}

extern "C" void kernel_launch(void* const* d_in, const int* in_sizes, int n_in,
                              void* d_out, int out_size, void* d_ws, size_t ws_size,
                              hipStream_t stream) {
    const float* pose = (const float*)d_in[0];  // x        [1,4,4]
    const float* mh   = (const float*)d_in[1];  // means_hom_tmp [1,4,768]
    const float* cov  = (const float*)d_in[2];  // cov_world [768,3,3]
    const float* opac = (const float*)d_in[3];  // opacities_rast [768]
    const float* Kmat = (const float*)d_in[4];  // K [3,3]
    // d_in[5] = tile_coord (recomputed arithmetically in-kernel)
    const float* limx = (const float*)d_in[6];
    const float* limy = (const float*)d_in[7];
    const float* fxp  = (const float*)d_in[8];
    const float* fyp  = (const float*)d_in[9];

    float* vtab = (float*)d_ws;                 // 8*768 floats = 24 KB

    splat_pre<<<(NG + 255) / 256, 256, 0, stream>>>(
        pose, mh, cov, opac, Kmat, limx, limy, fxp, fyp, vtab);

    splat_main<<<MAIN_BLOCKS, 256, 0, stream>>>(vtab, (float*)d_out);
}